// SAGE_34342558498881
// MI455X (gfx1250) — compile-verified
//
#include <hip/hip_runtime.h>

// SAGEConv x3 + linear classifier for MI455X (gfx1250, wave32, WMMA).
// Scatter phase is L2-resident (x: 25.6MB, agg: 25.6MB << 192MB L2) -> global
// f32 atomics. GEMM phase is compute-trivial (~10 GFLOP total) so we keep
// exact f32 numerics and use V_WMMA_F32_16X16X4_F32 on the matrix pipe.

typedef __attribute__((ext_vector_type(2))) float v2f;
typedef __attribute__((ext_vector_type(8))) float v8f;

#define NNODES 50000
#define NEDGES 1600000
#define DIN    128            // feature width entering every SAGE layer

// ---------------------------------------------------------------- degrees ---
__global__ void deg_kernel(const long long* __restrict__ dst,
                           float* __restrict__ deg, int E) {
    int e = blockIdx.x * blockDim.x + threadIdx.x;
    if (e < E) atomicAdd(&deg[(int)dst[e]], 1.0f);
}

// ------------------------------------------------------- gather+scatter-add --
// one thread per (edge, float4 chunk); 32 chunks cover the 128-wide row
__global__ void scatter_kernel(const float* __restrict__ xin,
                               const long long* __restrict__ src,
                               const long long* __restrict__ dst,
                               float* __restrict__ agg, int E) {
    int tid = blockIdx.x * blockDim.x + threadIdx.x;
    int e = tid >> 5;
    int c = tid & 31;
    if (e >= E) return;
    int s = (int)src[e];
    int d = (int)dst[e];
    float4 v = ((const float4*)xin)[(size_t)s * 32 + c];
    float* base = agg + (size_t)d * DIN + c * 4;
    atomicAdd(base + 0, v.x);
    atomicAdd(base + 1, v.y);
    atomicAdd(base + 2, v.z);
    atomicAdd(base + 3, v.w);
}

// ------------------------------------------- mean@Wl + x@Wr + b, tanh (WMMA) --
// grid.x = NNODES/16 row tiles (exact); one wave32 per 16x16 output tile.
// dOut = 128 -> 8 waves (256 thr), dOut = 64 -> 4 waves (128 thr).
__global__ void sage_gemm_tanh(const float* __restrict__ agg,
                               const float* __restrict__ deg,
                               const float* __restrict__ xin,
                               const float* __restrict__ Wl,
                               const float* __restrict__ Wr,
                               const float* __restrict__ bias,
                               float* __restrict__ out, int dOut) {
    __shared__ float sMean[16 * DIN];   // 8 KB
    __shared__ float sX[16 * DIN];      // 8 KB

    const int row0 = blockIdx.x * 16;
    const int tid  = threadIdx.x;

    // Stage 16 rows of mean (agg * 1/max(deg,1)) and x into LDS, float4-wide.
    const float4* av = (const float4*)(agg + (size_t)row0 * DIN);
    const float4* xv = (const float4*)(xin + (size_t)row0 * DIN);
    for (int i = tid; i < 16 * (DIN / 4); i += blockDim.x) {
        int r = i >> 5;                                  // DIN/4 == 32
        float inv = 1.0f / fmaxf(deg[row0 + r], 1.0f);
        float4 a  = av[i];
        float4 xx = xv[i];
        a.x *= inv; a.y *= inv; a.z *= inv; a.w *= inv;
        ((float4*)sMean)[i] = a;
        ((float4*)sX)[i]    = xx;
    }
    __syncthreads();

    const int wave = tid >> 5;
    const int lane = tid & 31;
    const int g    = lane >> 4;       // lane group 0/1
    const int lr   = lane & 15;
    const int n    = wave * 16 + lr;  // output column

    // V_WMMA_F32_16X16X4_F32 A layout: lanes 0-15 hold K=k0..k0+1 of row M=lr,
    // lanes 16-31 hold K=k0+2..k0+3. B mirrors it over columns.
    v8f acc = {};
    const float2* sm2 = (const float2*)sMean;
    const float2* sx2 = (const float2*)sX;
#pragma unroll 8
    for (int k0 = 0; k0 < DIN; k0 += 4) {
        const int ka = k0 + 2 * g;                       // even -> float2 ok
        float2 am = sm2[(lr * DIN + ka) >> 1];
        float2 ax = sx2[(lr * DIN + ka) >> 1];
        v2f Am = { am.x, am.y };
        v2f Ax = { ax.x, ax.y };
        v2f Bl = { Wl[(size_t)ka * dOut + n], Wl[(size_t)(ka + 1) * dOut + n] };
        v2f Br = { Wr[(size_t)ka * dOut + n], Wr[(size_t)(ka + 1) * dOut + n] };
        acc = __builtin_amdgcn_wmma_f32_16x16x4_f32(false, Am, false, Bl,
                                                    (short)0, acc, false, false);
        acc = __builtin_amdgcn_wmma_f32_16x16x4_f32(false, Ax, false, Br,
                                                    (short)0, acc, false, false);
    }

    // C/D layout: VGPR v -> M = v + 8*g, N = lr
    const float bv = bias[n];
#pragma unroll
    for (int v = 0; v < 8; ++v) {
        int m = v + 8 * g;
        out[(size_t)(row0 + m) * dOut + n] = tanhf(acc[v] + bv);
    }
}

// ------------------------------------------------------- classifier (64->8) --
__global__ void classifier_kernel(const float* __restrict__ h3,
                                  const float* __restrict__ Wc,
                                  const float* __restrict__ bc,
                                  float* __restrict__ out, int N) {
    int nid = blockIdx.x * blockDim.x + threadIdx.x;
    if (nid >= N) return;
    float acc[8];
#pragma unroll
    for (int c = 0; c < 8; ++c) acc[c] = bc[c];
    const float* row = h3 + (size_t)nid * 64;
#pragma unroll 4
    for (int k = 0; k < 64; ++k) {
        float hv = row[k];
#pragma unroll
        for (int c = 0; c < 8; ++c) acc[c] = fmaf(hv, Wc[k * 8 + c], acc[c]);
    }
    float4* o = (float4*)(out + (size_t)nid * 8);
    o[0] = make_float4(acc[0], acc[1], acc[2], acc[3]);
    o[1] = make_float4(acc[4], acc[5], acc[6], acc[7]);
}

// --------------------------------------------------------------------------- 
extern "C" void kernel_launch(void* const* d_in, const int* in_sizes, int n_in,
                              void* d_out, int out_size, void* d_ws, size_t ws_size,
                              hipStream_t stream) {
    const float*      x   = (const float*)d_in[0];
    const long long*  ei  = (const long long*)d_in[1];   // int64 edge_index
    const float*      W1l = (const float*)d_in[2];
    const float*      W1r = (const float*)d_in[3];
    const float*      b1  = (const float*)d_in[4];
    const float*      W2l = (const float*)d_in[5];
    const float*      W2r = (const float*)d_in[6];
    const float*      b2  = (const float*)d_in[7];
    const float*      W3l = (const float*)d_in[8];
    const float*      W3r = (const float*)d_in[9];
    const float*      b3  = (const float*)d_in[10];
    const float*      Wc  = (const float*)d_in[11];
    const float*      bc  = (const float*)d_in[12];

    const long long* src = ei;            // edge_index[0]
    const long long* dst = ei + NEDGES;   // edge_index[1]

    float* out = (float*)d_out;           // (h1, h2, h3, logits) concatenated
    float* h1 = out;
    float* h2 = h1 + (size_t)NNODES * 128;
    float* h3 = h2 + (size_t)NNODES * 128;
    float* lg = h3 + (size_t)NNODES * 64;

    float* agg = (float*)d_ws;                       // 25.6 MB
    float* deg = agg + (size_t)NNODES * 128;         // 0.2 MB

    const int scatterBlocks = (NEDGES * 32) / 256;   // 200000, exact
    const int rowTiles      = NNODES / 16;           // 3125, exact

    // degrees (same for every layer)
    hipMemsetAsync(deg, 0, NNODES * sizeof(float), stream);
    deg_kernel<<<(NEDGES + 255) / 256, 256, 0, stream>>>(dst, deg, NEDGES);

    // layer 1: x -> h1
    hipMemsetAsync(agg, 0, (size_t)NNODES * 128 * sizeof(float), stream);
    scatter_kernel<<<scatterBlocks, 256, 0, stream>>>(x, src, dst, agg, NEDGES);
    sage_gemm_tanh<<<rowTiles, 256, 0, stream>>>(agg, deg, x, W1l, W1r, b1, h1, 128);

    // layer 2: h1 -> h2
    hipMemsetAsync(agg, 0, (size_t)NNODES * 128 * sizeof(float), stream);
    scatter_kernel<<<scatterBlocks, 256, 0, stream>>>(h1, src, dst, agg, NEDGES);
    sage_gemm_tanh<<<rowTiles, 256, 0, stream>>>(agg, deg, h1, W2l, W2r, b2, h2, 128);

    // layer 3: h2 -> h3 (dOut = 64 -> 4 waves per block)
    hipMemsetAsync(agg, 0, (size_t)NNODES * 128 * sizeof(float), stream);
    scatter_kernel<<<scatterBlocks, 256, 0, stream>>>(h2, src, dst, agg, NEDGES);
    sage_gemm_tanh<<<rowTiles, 128, 0, stream>>>(agg, deg, h2, W3l, W3r, b3, h3, 64);

    // classifier: h3 -> logits
    classifier_kernel<<<(NNODES + 255) / 256, 256, 0, stream>>>(h3, Wc, bc, lg, NNODES);
}